// CapacityBasedRouter_24257975287909
// MI455X (gfx1250) — compile-verified
//
#include <hip/hip_runtime.h>

#define D_MODEL     4096
#define N_TOKENS    8192
#define NUM_EXPERTS 8
#define CAPACITY    1024           // N_TOKENS / NUM_EXPERTS * 1.0
#define N_SLOTS     (N_TOKENS * 2) // TOP_K = 2

typedef __attribute__((ext_vector_type(2))) float v2f;
typedef __attribute__((ext_vector_type(4))) float v4f;
typedef __attribute__((ext_vector_type(8))) float v8f;
typedef unsigned long long u64;

// ---- workspace layout (float units) ----
// [0     , 65536) logits  (8192 x 8)
// [65536 , 98304) keys (u64[16384]): e<<46 | pbits<<14 | (16383 - slot)
// [98304 , 98312) colsum (8)
// [98312]         zsum
// [98313 , 98321) usage  (8)
#define WS_LOGITS 0
#define WS_KEYS   65536
#define WS_COLSUM 98304
#define WS_ZSUM   98312
#define WS_USAGE  98313

__global__ void moe_init(float* __restrict__ w) {
    int t = threadIdx.x;
    if (t < 17) w[WS_COLSUM + t] = 0.0f; // colsum[8] + zsum[1] + usage[8]
}

// ---------------------------------------------------------------------------
// Kernel 1: logits = x @ gw^T via V_WMMA_F32_16X16X4_F32 (wave32 fp32 WMMA).
// 128 threads = 4 waves; each wave owns a 16-token x 16-expert tile
// (expert cols 8..15 read a dedicated zero row -> no EXEC-masked LDS loads).
// K swept in 128-wide chunks; gw^T chunk staged as b128 loads/stores.
// ---------------------------------------------------------------------------
#define KC      128
#define SGW_LD  132  // row stride (floats): 16B-aligned, +4 bank skew
__global__ __launch_bounds__(128) void moe_logits_wmma(
        const float* __restrict__ x, const float* __restrict__ gw,
        float* __restrict__ logits) {
    __shared__ float sgwT[9][SGW_LD]; // rows 0..7: gw chunk; row 8: zeros

    const int tid    = threadIdx.x;
    const int lane   = tid & 31;
    const int wave   = tid >> 5;
    const int col16  = lane & 15;            // N column / A row selector
    const int khalf  = lane >> 4;            // K pair select within step
    const int rowsel = min(col16, NUM_EXPERTS); // cols >=8 -> zero row
    const int tok0   = blockIdx.x * 64 + wave * 16;

    const float* xrow = x + (size_t)(tok0 + col16) * D_MODEL; // A row

    // zero row 8 once (33 float4 = 132 floats)
    if (tid < SGW_LD / 4) {
        v4f z = {};
        *(v4f*)&sgwT[NUM_EXPERTS][tid * 4] = z;
    }

    // staging indices: 8 rows x 128 floats = 256 float4, 2 per thread
    const int se0 = tid >> 5, sk0 = (tid & 31) * 4;        // idx = tid
    const int se1 = (tid + 128) >> 5, sk1 = sk0;           // idx = tid + 128

    v8f c = {};
    for (int kb = 0; kb < D_MODEL; kb += KC) {
        v4f g0 = *(const v4f*)(gw + se0 * D_MODEL + kb + sk0);
        v4f g1 = *(const v4f*)(gw + se1 * D_MODEL + kb + sk1);
        __syncthreads();                     // protect prior chunk's readers
        *(v4f*)&sgwT[se0][sk0] = g0;
        *(v4f*)&sgwT[se1][sk1] = g1;
        __syncthreads();

        #pragma unroll
        for (int ks = 0; ks < KC; ks += 4) {
            const int kl = ks + 2 * khalf;
            // A fragment: x[M=col16][K=kl, kl+1]
            v2f a = *(const v2f*)(xrow + kb + kl);
            // B fragment: gw^T[K=kl, kl+1][N=col16] (zeros for col16 >= 8)
            v2f b = *(const v2f*)&sgwT[rowsel][kl];
            c = __builtin_amdgcn_wmma_f32_16x16x4_f32(
                    false, a, false, b, (short)0, c, false, false);
        }
    }

    // C layout: VGPR r -> M=r (lanes 0-15) / M=r+8 (lanes 16-31), N=col16
    #pragma unroll
    for (int r = 0; r < 8; ++r) {
        int token = tok0 + r + (khalf ? 8 : 0);
        if (col16 < NUM_EXPERTS)
            logits[token * NUM_EXPERTS + col16] = c[r];
    }
}

// ---------------------------------------------------------------------------
// Kernel 2: per-token softmax, top-2, sortable slot keys, loss partials
// ---------------------------------------------------------------------------
__global__ __launch_bounds__(256) void moe_route(
        const float* __restrict__ logits, u64* __restrict__ keys,
        float* __restrict__ colsum, float* __restrict__ zsum) {
    __shared__ float s_col[NUM_EXPERTS];
    __shared__ float s_z;
    const int tid = threadIdx.x;
    if (tid < NUM_EXPERTS) s_col[tid] = 0.0f;
    if (tid == 0) s_z = 0.0f;
    __syncthreads();

    const int t = blockIdx.x * 256 + tid;
    float l[NUM_EXPERTS];
    #pragma unroll
    for (int e = 0; e < NUM_EXPERTS; ++e) l[e] = logits[t * NUM_EXPERTS + e];

    float m = l[0];
    #pragma unroll
    for (int e = 1; e < NUM_EXPERTS; ++e) m = fmaxf(m, l[e]);
    float s = 0.0f, p[NUM_EXPERTS];
    #pragma unroll
    for (int e = 0; e < NUM_EXPERTS; ++e) { p[e] = __expf(l[e] - m); s += p[e]; }
    const float inv = 1.0f / s;
    #pragma unroll
    for (int e = 0; e < NUM_EXPERTS; ++e) atomicAdd(&s_col[e], p[e] * inv);
    const float lse = m + __logf(s);
    atomicAdd(&s_z, lse * lse);

    // top-2, first-occurrence tie-break (strict >), matching lax.top_k
    float b1 = -INFINITY, b2 = -INFINITY;
    int   i1 = 0, i2 = 0;
    #pragma unroll
    for (int e = 0; e < NUM_EXPERTS; ++e) {
        if (l[e] > b1)      { b2 = b1; i2 = i1; b1 = l[e]; i1 = e; }
        else if (l[e] > b2) { b2 = l[e]; i2 = e; }
    }
    const float t2 = __expf(b2 - b1);       // <= 1
    const float d  = 1.0f / (1.0f + t2);
    const float p1 = d;                     // in [0.5, 1]
    const float p2 = t2 * d;                // 1 - p1

    // key = expert<<46 | prob_bits<<14 | (16383 - slot); positive-float bit
    // patterns are order-preserving, so u64 order == routing priority.
    const int s0 = t * 2, s1 = t * 2 + 1;
    keys[s0] = ((u64)i1 << 46) | ((u64)__float_as_uint(p1) << 14)
             | (u64)(N_SLOTS - 1 - s0);
    keys[s1] = ((u64)i2 << 46) | ((u64)__float_as_uint(p2) << 14)
             | (u64)(N_SLOTS - 1 - s1);

    __syncthreads();
    if (tid < NUM_EXPERTS) atomicAdd(&colsum[tid], s_col[tid]);
    if (tid == 0) atomicAdd(zsum, s_z);
}

// ---------------------------------------------------------------------------
// Kernel 3: capacity ranking via all-pairs count on packed keys:
// rank(i) = #{j : same expert && key_j > key_i}  (== stable double-argsort)
// ---------------------------------------------------------------------------
#define RCHUNK 4096
__global__ __launch_bounds__(256) void moe_rank(
        const u64* __restrict__ keys, float* __restrict__ out_idx,
        float* __restrict__ out_p, float* __restrict__ usage) {
    __shared__ u64 skey[RCHUNK]; // 32 KB
    const int tid = threadIdx.x;
    const int i   = blockIdx.x * 256 + tid;

    const u64 my_key = keys[i];
    int cnt = 0;
    for (int cb = 0; cb < N_SLOTS; cb += RCHUNK) {
        #pragma unroll
        for (int j = 0; j < RCHUNK / 256; ++j)
            skey[tid + j * 256] = keys[cb + tid + j * 256];
        __syncthreads();
        #pragma unroll 8
        for (int j = 0; j < RCHUNK; ++j) {
            const u64 k = skey[j];
            const bool beats = (((k ^ my_key) >> 46) == 0) && (k > my_key);
            cnt += beats ? 1 : 0;
        }
        __syncthreads();
    }

    const unsigned int my_e = (unsigned int)(my_key >> 46);
    const float        my_p = __uint_as_float((unsigned int)(my_key >> 14));
    const bool keep = (cnt < CAPACITY);
    out_idx[i] = keep ? (float)my_e : -1.0f;
    out_p[i]   = keep ? my_p : 0.0f;
    // usage counts kept top-1 assignments (dropped top1 == -1 contributes 0)
    if (((i & 1) == 0) && keep) atomicAdd(&usage[my_e], 1.0f);
}

// ---------------------------------------------------------------------------
// Kernel 4: finalize scalar losses + copy usage
// ---------------------------------------------------------------------------
__global__ void moe_final(const float* __restrict__ colsum,
                          const float* __restrict__ zsum,
                          const float* __restrict__ usage,
                          float* __restrict__ out) {
    if (threadIdx.x == 0) {
        float lb = 0.0f;
        for (int e = 0; e < NUM_EXPERTS; ++e) lb += colsum[e] * usage[e];
        out[2 * N_SLOTS]     = lb * (0.01f / ((float)N_TOKENS * NUM_EXPERTS));
        out[2 * N_SLOTS + 1] = (*zsum / (float)N_TOKENS) * 0.001f;
        for (int e = 0; e < NUM_EXPERTS; ++e)
            out[2 * N_SLOTS + 2 + e] = usage[e];
    }
}

extern "C" void kernel_launch(void* const* d_in, const int* in_sizes, int n_in,
                              void* d_out, int out_size, void* d_ws, size_t ws_size,
                              hipStream_t stream) {
    const float* x  = (const float*)d_in[0];  // (8192, 4096) f32
    const float* gw = (const float*)d_in[1];  // (8, 4096) f32
    float* out = (float*)d_out;               // 32778 floats
    float* w   = (float*)d_ws;

    float* logits = w + WS_LOGITS;
    u64*   keys   = (u64*)(w + WS_KEYS);
    float* colsum = w + WS_COLSUM;
    float* zsum   = w + WS_ZSUM;
    float* usage  = w + WS_USAGE;

    moe_init<<<1, 32, 0, stream>>>(w);
    moe_logits_wmma<<<N_TOKENS / 64, 128, 0, stream>>>(x, gw, logits);
    moe_route<<<N_TOKENS / 256, 256, 0, stream>>>(logits, keys, colsum, zsum);
    moe_rank<<<N_SLOTS / 256, 256, 0, stream>>>(keys, out, out + N_SLOTS, usage);
    moe_final<<<1, 1, 0, stream>>>(colsum, zsum, usage, out);
}